// AXSEmbeddingV2_74852690034821
// MI455X (gfx1250) — compile-verified
//
#include <hip/hip_runtime.h>

typedef float v4f __attribute__((ext_vector_type(4)));

#define TOK_PER_BLOCK 8

// ds_swizzle group-of-32 encoding: offset[4:0]=and_mask, [9:5]=or_mask, [14:10]=xor_mask
#define SWZ(x, IMM) __int_as_float(__builtin_amdgcn_ds_swizzle(__float_as_int(x), (IMM)))
#define TOP2_MERGE(IMM)                                              \
  {                                                                  \
    float o1 = SWZ(m1, IMM), o2 = SWZ(m2, IMM);                      \
    m2 = fmaxf(fminf(m1, o1), fmaxf(m2, o2));                        \
    m1 = fmaxf(m1, o1);                                              \
  }

// Acklam's rational approximation of the inverse normal CDF (ndtri).
__device__ __forceinline__ float ndtri_acklam(float p) {
  const float a1 = -3.969683028665376e+01f, a2 = 2.209460984245205e+02f,
              a3 = -2.759285104469687e+02f, a4 = 1.383577518672690e+02f,
              a5 = -3.066479806614716e+01f, a6 = 2.506628277459239e+00f;
  const float b1 = -5.447609879822406e+01f, b2 = 1.615858368580409e+02f,
              b3 = -1.556989798598866e+02f, b4 = 6.680131188771972e+01f,
              b5 = -1.328068155288572e+01f;
  const float c1 = -7.784894002430293e-03f, c2 = -3.223964580411365e-01f,
              c3 = -2.400758277161838e+00f, c4 = -2.549732539343734e+00f,
              c5 = 4.374664141464968e+00f,  c6 = 2.938163982698783e+00f;
  const float d1 = 7.784695709041462e-03f, d2 = 3.224671290700398e-01f,
              d3 = 2.445134137142996e+00f, d4 = 3.754408661907416e+00f;
  const float plow = 0.02425f;
  if (p < plow) {
    float q = sqrtf(-2.0f * logf(p));
    return (((((c1 * q + c2) * q + c3) * q + c4) * q + c5) * q + c6) /
           ((((d1 * q + d2) * q + d3) * q + d4) * q + 1.0f);
  } else if (p <= 1.0f - plow) {
    float q = p - 0.5f, r = q * q;
    return (((((a1 * r + a2) * r + a3) * r + a4) * r + a5) * r + a6) * q /
           (((((b1 * r + b2) * r + b3) * r + b4) * r + b5) * r + 1.0f);
  } else {
    float q = sqrtf(-2.0f * logf(1.0f - p));
    return -(((((c1 * q + c2) * q + c3) * q + c4) * q + c5) * q + c6) /
           ((((d1 * q + d2) * q + d3) * q + d4) * q + 1.0f);
  }
}

// One workgroup (256 thr = 8 waves) handles TOK_PER_BLOCK token rows of D floats.
// Within a row pass: each 16-lane half-group owns one 64-element quant block
// (4 floats/lane via one b128 load). Next row is software-prefetched while the
// current row is quantized (global_prefetch_b8, fire-and-forget).
__global__ __launch_bounds__(256, 8) void AXSEmbeddingV2_nf5_kernel(
    const float* __restrict__ weight, const int* __restrict__ tok,
    float* __restrict__ out, int D, int tokens) {
  __shared__ float s_mids[32];  // 31 midpoints used
  __shared__ float s_lvls[32];  // 32 NF5 levels, normalized to [-1,1]

  const int tid = threadIdx.x;

  // --- Build NF5 codebook in LDS once per block (amortized over 8 rows) ---
  float raw = 0.0f;
  if (tid < 32) {
    raw = ndtri_acklam((2.0f * (float)tid + 1.0f) * 0.015625f);  // (2k+1)/64
    s_lvls[tid] = raw;
  }
  __syncthreads();
  const float norm = fmaxf(fabsf(s_lvls[0]), fabsf(s_lvls[31]));
  const float rnext = (tid < 31) ? s_lvls[tid + 1] : 0.0f;
  __syncthreads();
  if (tid < 32) {
    s_lvls[tid] = raw / norm;
    if (tid < 31) s_mids[tid] = 0.5f * (raw + rnext) / norm;
  }
  __syncthreads();

  const int t0 = blockIdx.x * TOK_PER_BLOCK;

#pragma unroll 1
  for (int i = 0; i < TOK_PER_BLOCK; ++i) {
    const int token = t0 + i;
    if (token >= tokens) break;

    // wave-uniform index -> scalar load path
    const float* __restrict__ src = weight + (size_t)tok[token] * (size_t)D;
    float* __restrict__ dst = out + (size_t)token * (size_t)D;

    // Prefetch the next gathered row's cachelines (one address per lane
    // covering exactly what we will demand-load next iteration) so the
    // random-row HBM fetch overlaps with this row's quantization math.
    if (i + 1 < TOK_PER_BLOCK && token + 1 < tokens) {
      const float* __restrict__ nsrc =
          weight + (size_t)tok[token + 1] * (size_t)D;
      __builtin_prefetch(nsrc + tid * 4, 0, 3);  // -> global_prefetch_b8
    }

#pragma unroll 1
    for (int off = tid * 4; off < D; off += 256 * 4) {
      const v4f v = *(const v4f*)(src + off);

      // local top-2 of |v| over the lane's 4 elements
      float a0 = fabsf(v.x), a1 = fabsf(v.y), a2 = fabsf(v.z), a3 = fabsf(v.w);
      float m1 = fmaxf(a0, a1), m2 = fminf(a0, a1);
      m2 = fmaxf(m2, fminf(m1, a2)); m1 = fmaxf(m1, a2);
      m2 = fmaxf(m2, fminf(m1, a3)); m1 = fmaxf(m1, a3);

      // 4-step XOR butterfly over the 16-lane half-group -> top-2 of 64 values
      TOP2_MERGE(0x041f)  // xor 1
      TOP2_MERGE(0x081f)  // xor 2
      TOP2_MERGE(0x101f)  // xor 4
      TOP2_MERGE(0x201f)  // xor 8

      // quantile(|x|, 0.999) over 64 samples == a[62] + 0.937*(a[63]-a[62])
      float scale = m2 + 0.937f * (m1 - m2);
      scale = fmaxf(scale, 1e-8f);
      const float inv = 1.0f / scale;

      // clamp, normalize, branchless lower_bound over 31 midpoints, dequant
      auto dq = [&](float x) -> float {
        const float xc = fminf(fmaxf(x, -scale), scale);
        const float t = xc * inv;
        int j = (s_mids[15] < t) ? 16 : 0;
        j += (s_mids[j + 7] < t) ? 8 : 0;
        j += (s_mids[j + 3] < t) ? 4 : 0;
        j += (s_mids[j + 1] < t) ? 2 : 0;
        j += (s_mids[j] < t) ? 1 : 0;  // j in [0,31]
        return s_lvls[j] * scale;
      };

      v4f r;
      r.x = dq(v.x);
      r.y = dq(v.y);
      r.z = dq(v.z);
      r.w = dq(v.w);

      // 128MB streaming output: non-temporal so weight rows stay resident in L2
      __builtin_nontemporal_store(r, (v4f*)(dst + off));
    }
  }
}

extern "C" void kernel_launch(void* const* d_in, const int* in_sizes, int n_in,
                              void* d_out, int out_size, void* d_ws, size_t ws_size,
                              hipStream_t stream) {
  const int* tok = (const int*)d_in[0];         // input  [8,4096] int32
  const float* weight = (const float*)d_in[1];  // weight [50257,1024] fp32
  float* out = (float*)d_out;                   // [8,4096,1024] fp32

  const int tokens = in_sizes[0];               // 32768
  const int D = out_size / tokens;              // 1024 (multiple of 128)

  dim3 grid((tokens + TOK_PER_BLOCK - 1) / TOK_PER_BLOCK), block(256);
  AXSEmbeddingV2_nf5_kernel<<<grid, block, 0, stream>>>(weight, tok, out, D,
                                                        tokens);
}